// TopKContSparsifier_58394375357263
// MI455X (gfx1250) — compile-verified
//
#include <hip/hip_runtime.h>

typedef float    v4f __attribute__((ext_vector_type(4)));
typedef int      v4i __attribute__((ext_vector_type(4)));
typedef unsigned v4u __attribute__((ext_vector_type(4)));
typedef unsigned u32;

#define N_COLS          4096
#define KTH             65          // v65 = 65th largest = sorted_asc[4031]
#define FRAC            0.953125f   // q*(n-1) - floor = 4030.953125 - 4030
#define ROWS_PER_BLOCK  8
#define THREADS         256

// gfx1250 async global->LDS copy (16B per lane), tracked by ASYNCcnt.
// NT hint: the input is a one-shot 256MB stream; don't thrash the 192MB L2.
__device__ __forceinline__ void async_ld16(const float* g, u32 lds_off) {
    asm volatile("global_load_async_to_lds_b128 %0, %1, off th:TH_LOAD_NT"
                 :: "v"(lds_off), "v"(g) : "memory");
}

__global__ __launch_bounds__(THREADS)
void topk_soft_threshold(const float* __restrict__ X,
                         float* __restrict__ out,
                         int nrows) {
    __shared__ __align__(16) float buf[2][N_COLS];   // 32KB double buffer
    __shared__ __align__(16) int   red[2][8];        // parity-buffered count partials
    __shared__ __align__(16) int   redc[8];
    __shared__ __align__(16) u32   redm[8];

    const int tid  = threadIdx.x;
    const int lane = tid & 31;
    const int wid  = tid >> 5;
    const int row0 = blockIdx.x * ROWS_PER_BLOCK;
    const int rend = min(ROWS_PER_BLOCK, nrows - row0);

    // Per-thread LDS byte offsets for its 4 disjoint 16B chunks per buffer.
    // (generic LDS pointer: low 32 bits == LDS byte offset on gfx1250)
    u32 ldsoff[2][4];
#pragma unroll
    for (int b = 0; b < 2; ++b)
#pragma unroll
        for (int j = 0; j < 4; ++j)
            ldsoff[b][j] = (u32)(size_t)&buf[b][j * 1024 + tid * 4];

    // Prime the pipeline: row0 -> buffer 0 (async, consumes no VGPRs).
    {
        const float* g = X + (size_t)row0 * N_COLS + tid * 4;
#pragma unroll
        for (int j = 0; j < 4; ++j) async_ld16(g + j * 1024, ldsoff[0][j]);
    }

    for (int r = 0; r < rend; ++r) {
        const int row = row0 + r;
        const int cb  = r & 1;

        if (r + 1 < rend) {
            // Drain our ds reads of buf[cb^1] (consumed last row) before the
            // async engine overwrites that buffer.
            asm volatile("s_wait_dscnt 0" ::: "memory");
            const float* g = X + (size_t)(row + 1) * N_COLS + tid * 4;
#pragma unroll
            for (int j = 0; j < 4; ++j) async_ld16(g + j * 1024, ldsoff[cb ^ 1][j]);
            // 8 async ops in flight; async loads complete in order, so
            // <=4 outstanding => current row's data has landed in LDS.
            asm volatile("s_wait_asynccnt 4" ::: "memory");
        } else {
            asm volatile("s_wait_asynccnt 0" ::: "memory");
        }

        // Pull this thread's 16 values from LDS; keep |x| bits + sign mask.
        // Each thread reads exactly the region it async-wrote: no barriers
        // needed anywhere in the data path.
        u32 a[16];
        u32 sgn = 0;
#pragma unroll
        for (int j = 0; j < 4; ++j) {
            v4f v = *(const v4f*)&buf[cb][j * 1024 + tid * 4];
#pragma unroll
            for (int i = 0; i < 4; ++i) {
                u32 u = __float_as_uint(v[i]);
                a[j * 4 + i] = u & 0x7fffffffu;
                sgn |= (u >> 31) << (j * 4 + i);
            }
        }

        // --- MSB-first greedy radix select: T = bits of 65th-largest |x| ---
        // (non-negative floats order as u32; count via ballot+popcount so the
        //  wave count lives in SGPRs: 1 VALU cmp per element, SALU co-issues)
        u32 T = 0;
#pragma unroll 1
        for (int bit = 30; bit >= 0; --bit) {
            const u32 cand = T | (1u << bit);
            int c = 0;
#pragma unroll
            for (int j = 0; j < 16; ++j)
                c += (int)__popcll(__ballot(a[j] >= cand));   // v_cmp + s_bcnt
            const int p = bit & 1;
            if (lane == 0) red[p][wid] = c;                   // c is wave-uniform
            __syncthreads();                                  // 1 barrier/bit
            v4i t0 = *(const v4i*)&red[p][0];
            v4i t1 = *(const v4i*)&red[p][4];
            int total = t0.x + t0.y + t0.z + t0.w + t1.x + t1.y + t1.z + t1.w;
            if (total >= KTH) T = cand;
        }

        // --- tie count and runner-up max (v66) ---
        int cge = 0;        // count(|x| >= v65): wave-uniform via ballot
        u32 m   = 0;        // max{|x| : |x| < v65}
#pragma unroll
        for (int j = 0; j < 16; ++j) {
            cge += (int)__popcll(__ballot(a[j] >= T));
            if (a[j] < T && a[j] > m) m = a[j];
        }
#pragma unroll
        for (int off = 16; off >= 1; off >>= 1) {
            u32 om = __shfl_xor(m, off, 32);
            m = om > m ? om : m;
        }
        if (lane == 0) { redc[wid] = cge; redm[wid] = m; }
        __syncthreads();
        v4i c0 = *(const v4i*)&redc[0];
        v4i c1 = *(const v4i*)&redc[4];
        int ctot = c0.x + c0.y + c0.z + c0.w + c1.x + c1.y + c1.z + c1.w;
        v4u m0 = *(const v4u*)&redm[0];
        v4u m1 = *(const v4u*)&redm[4];
        u32 mt = m0.x;
        mt = m0.y > mt ? m0.y : mt; mt = m0.z > mt ? m0.z : mt;
        mt = m0.w > mt ? m0.w : mt; mt = m1.x > mt ? m1.x : mt;
        mt = m1.y > mt ? m1.y : mt; mt = m1.z > mt ? m1.z : mt;
        mt = m1.w > mt ? m1.w : mt;
        __syncthreads();    // protects red/redc/redm reuse next row

        const float v65 = __uint_as_float(T);
        // If >=66 values are >= v65, the 66th largest is also v65 (ties).
        const float v66 = (ctot >= KTH + 1) ? v65 : __uint_as_float(mt);
        const float Q   = v66 + FRAC * (v65 - v66);

        // --- soft-threshold, restore sign, stream out (non-temporal) ---
        float* orow = out + (size_t)row * N_COLS + tid * 4;
#pragma unroll
        for (int j = 0; j < 4; ++j) {
            v4f o;
#pragma unroll
            for (int i = 0; i < 4; ++i) {
                const int k = j * 4 + i;
                float t = fmaxf(__uint_as_float(a[k]) - Q, 0.0f);
                o[i] = __uint_as_float(__float_as_uint(t) |
                                       (((sgn >> k) & 1u) << 31));
            }
            __builtin_nontemporal_store(o, (v4f*)(orow + j * 1024));
        }
    }
}

extern "C" void kernel_launch(void* const* d_in, const int* in_sizes, int n_in,
                              void* d_out, int out_size, void* d_ws, size_t ws_size,
                              hipStream_t stream) {
    const float* X  = (const float*)d_in[0];
    float* out      = (float*)d_out;
    const int nrows = in_sizes[0] / N_COLS;   // 16384
    const int grid  = (nrows + ROWS_PER_BLOCK - 1) / ROWS_PER_BLOCK;
    topk_soft_threshold<<<grid, THREADS, 0, stream>>>(X, out, nrows);
}